// DistributionLoss_6940667150680
// MI455X (gfx1250) — compile-verified
//
#include <hip/hip_runtime.h>

#ifndef __has_builtin
#define __has_builtin(x) 0
#endif

// Problem constants (reference setup: N=1e6, D=128, C=1000).
#define CC      1000        // num classes (compile-time: sizes LDS class table)
#define DD      128         // feature dim
#define CHUNK   32          // columns per block (128B per row -> perfect coalescing)
#define NCHUNK  4           // DD / CHUNK
#define TPB     256
#define RSTEP   32          // rows per inner step = TPB / (CHUNK/4)
#define MAXBUF  8           // max spread copies in atomic-flush fallback mode
#define CPOL_NT 1           // temporal hint NT: one-shot stream, don't churn L2

#define TRY_ASYNC 1
#if TRY_ASYNC && __has_builtin(__builtin_amdgcn_global_load_async_to_lds_b128) && \
    __has_builtin(__builtin_amdgcn_s_wait_asynccnt)
#define USE_ASYNC_LD 1
#else
#define USE_ASYNC_LD 0
#endif
#if TRY_ASYNC && __has_builtin(__builtin_amdgcn_global_store_async_from_lds_b128) && \
    __has_builtin(__builtin_amdgcn_s_wait_asynccnt)
#define USE_ASYNC_ST 1
#else
#define USE_ASYNC_ST 0
#endif

#if USE_ASYNC_LD || USE_ASYNC_ST
typedef int v4i_t __attribute__((ext_vector_type(4)));
typedef __attribute__((address_space(1))) v4i_t* g_v4i_p;
typedef __attribute__((address_space(3))) v4i_t* l_v4i_p;
template <int N_> __device__ __forceinline__ void wait_async() {
    __builtin_amdgcn_s_wait_asynccnt(N_);
}
#endif
#if USE_ASYNC_LD
__device__ __forceinline__ void async_g2l_b128(void* lds, const void* g) {
    __builtin_amdgcn_global_load_async_to_lds_b128(
        (g_v4i_p)g, (l_v4i_p)lds, /*offset=*/0, /*cpol=*/CPOL_NT);
}
#endif
#if USE_ASYNC_ST
__device__ __forceinline__ void async_l2g_b128(void* g, const void* lds) {
    __builtin_amdgcn_global_store_async_from_lds_b128(
        (g_v4i_p)g, (l_v4i_p)lds, /*offset=*/0, /*cpol=*/CPOL_NT);
}
#endif

// One pass over w1: per-block LDS class table (privatized segment-sum for a
// 32-column chunk), per-thread sum-of-squares, then one flush per block.
// percopy=1: each row-group owns a private global copy -> pure streaming
// stores (async LDS->global DMA), zero atomics. percopy=0: bucket-spread
// global atomic flush (small-workspace fallback).
__global__ __launch_bounds__(TPB, 1)
void seg_main(const float* __restrict__ w1, const int* __restrict__ Y,
              float* __restrict__ sums_buf, unsigned* __restrict__ counts,
              double* __restrict__ acc, int N, int rowsPerGroup,
              int ncopies, int percopy)
{
    __shared__ __align__(16) float cls[CC * CHUNK];  // 125 KB class-sum table
    __shared__ unsigned cnt[CC];                     // 4 KB histogram (chunk-0 blocks)
    __shared__ float4   stage[2][TPB];               // 8 KB double-buffered staging
    __shared__ float    red[TPB];

    const int t = threadIdx.x;
    const int j = blockIdx.x & (NCHUNK - 1);   // column chunk
    const int g = blockIdx.x >> 2;             // row group
    const int rowBeg = g * rowsPerGroup;
    const int rowEnd = min(N, rowBeg + rowsPerGroup);

    for (int i = t; i < CC * CHUNK; i += TPB) cls[i] = 0.0f;
    if (j == 0) for (int i = t; i < CC; i += TPB) cnt[i] = 0u;
    __syncthreads();

    const int rsub = t >> 3;   // 0..31: row within a 32-row batch
    const int c4   = t & 7;    // float4 slot within the 32-column chunk
    const float* gbase = w1 + (size_t)j * CHUNK + (size_t)c4 * 4;

    float sumsq = 0.0f;
    const int nbat = (rowEnd > rowBeg) ? ((rowEnd - rowBeg + RSTEP - 1) / RSTEP) : 0;

#if USE_ASYNC_LD
    // Every wave issues exactly one async b128 per batch (tail addresses are
    // clamped so EXEC stays full) -> per-wave ASYNCcnt math is exact: after
    // issuing batch b+1, waiting <=1 guarantees batch b landed, and each lane
    // only reads back the 16B slot it requested itself (no barrier needed).
    if (nbat > 0) {
        int r0 = rowBeg + rsub; if (r0 >= rowEnd) r0 = rowEnd - 1;
        async_g2l_b128(&stage[0][t], gbase + (size_t)r0 * DD);
    }
    for (int b = 0; b < nbat; ++b) {
        if (b + 1 < nbat) {
            int rn = rowBeg + (b + 1) * RSTEP + rsub; if (rn >= rowEnd) rn = rowEnd - 1;
            async_g2l_b128(&stage[(b + 1) & 1][t], gbase + (size_t)rn * DD);
            wait_async<1>();
        } else {
            wait_async<0>();
        }
        const int row = rowBeg + b * RSTEP + rsub;
        if (row < rowEnd) {
            const float4 v = stage[b & 1][t];
            sumsq += v.x * v.x + v.y * v.y + v.z * v.z + v.w * v.w;
            const int y = Y[row];
            float* d = &cls[y * CHUNK + (c4 << 2)];
            atomicAdd(d + 0, v.x);
            atomicAdd(d + 1, v.y);
            atomicAdd(d + 2, v.z);
            atomicAdd(d + 3, v.w);
            if (j == 0 && c4 == 0) atomicAdd(&cnt[y], 1u);
        }
    }
#else
    for (int b = 0; b < nbat; ++b) {
        const int row = rowBeg + b * RSTEP + rsub;
        if (row < rowEnd) {
            const float4 v = *(const float4*)(gbase + (size_t)row * DD);
            sumsq += v.x * v.x + v.y * v.y + v.z * v.z + v.w * v.w;
            const int y = Y[row];
            float* d = &cls[y * CHUNK + (c4 << 2)];
            atomicAdd(d + 0, v.x);
            atomicAdd(d + 1, v.y);
            atomicAdd(d + 2, v.z);
            atomicAdd(d + 3, v.w);
            if (j == 0 && c4 == 0) atomicAdd(&cnt[y], 1u);
        }
    }
#endif

    __syncthreads();

    if (percopy) {
        // Private copy per row-group: whole table streamed out, no atomics.
        float* dstb = sums_buf + (size_t)g * (CC * DD);
        for (int i4 = t; i4 < (CC * CHUNK) / 4; i4 += TPB) {
            const int i = i4 << 2;
            const int c = i >> 5;      // / CHUNK
            const int k = i & 31;      // % CHUNK
            float* dst = &dstb[(size_t)c * DD + j * CHUNK + k];
#if USE_ASYNC_ST
            async_l2g_b128(dst, &cls[i]);   // drains via ENDPGM implicit wait-idle
#else
            *(float4*)dst = *(const float4*)&cls[i];
#endif
        }
    } else {
        // Fallback: bucket-spread global atomics (L2-resident).
        float* dstb = sums_buf + (size_t)(g & (ncopies - 1)) * (CC * DD);
        for (int i = t; i < CC * CHUNK; i += TPB) {
            const float v = cls[i];
            if (v != 0.0f) {
                const int c = i >> 5;
                const int k = i & 31;
                atomicAdd(&dstb[(size_t)c * DD + j * CHUNK + k], v);
            }
        }
    }
    if (j == 0) {
        for (int i = t; i < CC; i += TPB) {
            const unsigned v = cnt[i];
            if (v) atomicAdd(&counts[i], v);
        }
    }

    // Block-reduce sum of squares; one f64 global atomic per block.
    red[t] = sumsq;
    __syncthreads();
    for (int s = TPB >> 1; s > 0; s >>= 1) {
        if (t < s) red[t] += red[t + s];
        __syncthreads();
    }
    if (t == 0) atomicAdd(&acc[0], (double)red[0]);
}

// Fold the copies and accumulate  sum_c ||s_c||^2 / max(n_c,1).
__global__ __launch_bounds__(TPB)
void cross_fold(const float* __restrict__ sums_buf, const unsigned* __restrict__ counts,
                double* __restrict__ acc, int ncopies)
{
    __shared__ float red[TPB];
    const int idx = blockIdx.x * TPB + threadIdx.x;
    float val = 0.0f;
    if (idx < CC * DD) {
        float s = 0.0f;
        for (int b = 0; b < ncopies; ++b) s += sums_buf[(size_t)b * (CC * DD) + idx];
        const unsigned n = counts[idx >> 7];           // idx / DD
        val = (s * s) / (float)(n > 1u ? n : 1u);
    }
    red[threadIdx.x] = val;
    __syncthreads();
    for (int s = TPB >> 1; s > 0; s >>= 1) {
        if (threadIdx.x < s) red[threadIdx.x] += red[threadIdx.x + s];
        __syncthreads();
    }
    if (threadIdx.x == 0) atomicAdd(&acc[1], (double)red[0]);
}

__global__ void finalize_k(const double* __restrict__ acc, float* __restrict__ out, int N)
{
    if (threadIdx.x == 0 && blockIdx.x == 0)
        out[0] = (float)((acc[0] - acc[1]) / (double)N);
}

extern "C" void kernel_launch(void* const* d_in, const int* in_sizes, int n_in,
                              void* d_out, int out_size, void* d_ws, size_t ws_size,
                              hipStream_t stream)
{
    const float* w1 = (const float*)d_in[0];
    const int*   Y  = (const int*)d_in[1];
    const int N = in_sizes[1];                 // 1e6; D = in_sizes[0]/N = 128

    char* ws = (char*)d_ws;
    double*   acc    = (double*)ws;                                  // acc[0]=sumsq, acc[1]=cross
    unsigned* counts = (unsigned*)(ws + 64);                         // CC u32
    const size_t sumsOff = (64 + (size_t)CC * 4 + 255) & ~(size_t)255;
    float* sums_buf = (float*)(ws + sumsOff);

    const int rowsPerGroup = 16384;            // 62 row groups x 4 chunks = 248 blocks
    const int G = (N + rowsPerGroup - 1) / rowsPerGroup;
    const size_t copyBytes = (size_t)CC * DD * sizeof(float);

    int percopy, ncopies;
    if (sumsOff + (size_t)G * copyBytes <= ws_size) {
        percopy = 1;                           // ~32 MB: private copy per group, no atomics
        ncopies = G;
    } else {
        percopy = 0;                           // atomic flush over nbuf spread copies
        ncopies = 1;
        while (ncopies < MAXBUF &&
               sumsOff + (size_t)(ncopies * 2) * copyBytes <= ws_size)
            ncopies <<= 1;
    }
    // percopy mode fully overwrites every copy -> only header needs zeroing.
    const size_t zeroBytes = percopy ? sumsOff : sumsOff + (size_t)ncopies * copyBytes;
    (void)hipMemsetAsync(d_ws, 0, zeroBytes, stream);

    seg_main<<<dim3(G * NCHUNK), dim3(TPB), 0, stream>>>(
        w1, Y, sums_buf, counts, acc, N, rowsPerGroup, ncopies, percopy);

    const int total = CC * DD;
    cross_fold<<<dim3((total + TPB - 1) / TPB), dim3(TPB), 0, stream>>>(
        sums_buf, counts, acc, ncopies);

    finalize_k<<<dim3(1), dim3(64), 0, stream>>>(acc, (float*)d_out, N);
}